// AttentionMechanism_21646635172225
// MI455X (gfx1250) — compile-verified
//
#include <hip/hip_runtime.h>

// Problem constants (from reference): x is (N=8, C=128, T=32, H=28, W=28) fp32.
#define NB   8
#define CC   128
#define TT   32
#define HW   784            // 28*28, row stride (in floats) of each 32x784 tile
#define TILE (TT * HW)      // floats per (n,c) tile

typedef __attribute__((ext_vector_type(2))) float v2f;
typedef __attribute__((ext_vector_type(8))) float v8f;

// D(16x16,f32) = A(16x4,f32) * B(4x16,f32) + C  -- full-precision CDNA5 WMMA
__device__ __forceinline__ v8f wmma4(v2f a, v2f b, v8f c) {
    return __builtin_amdgcn_wmma_f32_16x16x4_f32(
        /*neg_a=*/false, a, /*neg_b=*/false, b,
        /*c_mod=*/(short)0, c, /*reuse_a=*/false, /*reuse_b=*/false);
}

// ---------------------------------------------------------------------------
// Kernel 1: per-(n,c) Gram matrices, reduced per block -> partial[128][32][32]
// grid = 128 (n*16 + cgroup), block = 256 (8 waves, wave w -> c = cgroup*8+w)
// ---------------------------------------------------------------------------
__global__ __launch_bounds__(256) void gram_kernel(const float* __restrict__ x,
                                                   float* __restrict__ partial) {
    const int blk  = blockIdx.x;          // n*16 + g
    const int wave = threadIdx.x >> 5;
    const int lane = threadIdx.x & 31;
    const int half = lane >> 4;           // 0: k in {0,1}; 1: k in {2,3}
    const int lr   = lane & 15;

    // tile index = n*128 + c = blk*8 + wave
    const float* A  = x + (size_t)(blk * 8 + wave) * TILE;
    const float* p0 = A + (size_t)lr * HW + 2 * half;          // rows 0..15
    const float* p1 = A + (size_t)(16 + lr) * HW + 2 * half;   // rows 16..31

    v8f c00 = {}, c01 = {}, c10 = {}, c11 = {};
#pragma unroll 4
    for (int d = 0; d < HW; d += 4) {
        v2f a0 = *(const v2f*)(p0 + d);   // A-frag rows 0..15  (== B-frag of A^T)
        v2f a1 = *(const v2f*)(p1 + d);   // A-frag rows 16..31
        c00 = wmma4(a0, a0, c00);
        c01 = wmma4(a0, a1, c01);
        c10 = wmma4(a1, a0, c10);
        c11 = wmma4(a1, a1, c11);
    }

    __shared__ float lds[8][TT * TT];     // 32 KB
    {
        const int r0 = 8 * half;          // C layout: row = v + 8*half, col = lr
#pragma unroll
        for (int v = 0; v < 8; ++v) {
            lds[wave][(r0 + v) * 32 + lr]           = c00[v];
            lds[wave][(r0 + v) * 32 + 16 + lr]      = c01[v];
            lds[wave][(16 + r0 + v) * 32 + lr]      = c10[v];
            lds[wave][(16 + r0 + v) * 32 + 16 + lr] = c11[v];
        }
    }
    __syncthreads();

    // reduce the 8 per-wave grams -> one 32x32 partial per block
#pragma unroll
    for (int i = threadIdx.x; i < TT * TT; i += 256) {
        float s = 0.f;
#pragma unroll
        for (int w = 0; w < 8; ++w) s += lds[w][i];
        partial[(size_t)blk * (TT * TT) + i] = s;
    }
}

// ---------------------------------------------------------------------------
// Kernel 2: reduce 16 partials per n + row softmax.  grid = 8, block = 1024.
// wave t handles row t; lane s handles column s (wave32 == one 32-wide row).
// ---------------------------------------------------------------------------
__global__ __launch_bounds__(1024) void softmax_kernel(const float* __restrict__ partial,
                                                       float* __restrict__ attn) {
    const int n = blockIdx.x;
    const int t = threadIdx.x >> 5;
    const int s = threadIdx.x & 31;

    float e = 0.f;
#pragma unroll
    for (int g = 0; g < 16; ++g)
        e += partial[(size_t)((n * 16 + g) << 10) + (t << 5) + s];

    float m = e;
#pragma unroll
    for (int off = 16; off; off >>= 1) m = fmaxf(m, __shfl_xor(m, off, 32));
    float p = __expf(e - m);
    float sum = p;
#pragma unroll
    for (int off = 16; off; off >>= 1) sum += __shfl_xor(sum, off, 32);

    attn[(size_t)(n << 10) + (t << 5) + s] = p / sum;
}

// ---------------------------------------------------------------------------
// Kernel 3: out = alpha * (attn @ A) + A per (n,c) tile via WMMA.
// grid = 1024 (n*128+c), block = 256 (8 waves share the 49 column chunks).
// ---------------------------------------------------------------------------
__global__ __launch_bounds__(256) void apply_kernel(const float* __restrict__ x,
                                                    const float* __restrict__ attn,
                                                    const float* __restrict__ alphap,
                                                    float* __restrict__ out) {
    const int blk   = blockIdx.x;         // n*128 + c
    const int n     = blk >> 7;
    const float alpha = alphap[0];

    __shared__ float P[TT * TT];
    for (int i = threadIdx.x; i < TT * TT; i += 256) P[i] = attn[(size_t)(n << 10) + i];
    __syncthreads();

    const int wave = threadIdx.x >> 5;
    const int lane = threadIdx.x & 31;
    const int half = lane >> 4;
    const int lr   = lane & 15;

    // A-fragments of attn: pf[tb][kb] covers rows tb*16.., k = kb*4 + 2*half + {0,1}
    v2f pf[2][8];
#pragma unroll
    for (int tb = 0; tb < 2; ++tb)
#pragma unroll
        for (int kb = 0; kb < 8; ++kb) {
            const int row = tb * 16 + lr;
            const int k   = kb * 4 + 2 * half;
            v2f t; t.x = P[row * 32 + k]; t.y = P[row * 32 + k + 1];
            pf[tb][kb] = t;
        }

    const float* A = x   + (size_t)blk * TILE;
    float*       O = out + (size_t)blk * TILE;

    for (int chunk = wave; chunk < HW / 16; chunk += 8) {   // 49 chunks of 16 cols
        const int hw0 = chunk * 16;
        v8f c0 = {}, c1 = {};
        const float* bp = A + hw0 + lr + (size_t)(2 * half) * HW;
#pragma unroll
        for (int kb = 0; kb < 8; ++kb) {
            v2f b;
            b.x = bp[(size_t)(kb * 4) * HW];       // B[k][col] = A[s=k][hw0+col]
            b.y = bp[(size_t)(kb * 4 + 1) * HW];
            c0 = wmma4(pf[0][kb], b, c0);
            c1 = wmma4(pf[1][kb], b, c1);
        }
        const int col = hw0 + lr;
#pragma unroll
        for (int v = 0; v < 8; ++v) {
            const int r0 = v + 8 * half;           // rows 0..15 block
            O[(size_t)r0 * HW + col] = alpha * c0[v] + A[(size_t)r0 * HW + col];
            const int r1 = 16 + r0;                // rows 16..31 block
            O[(size_t)r1 * HW + col] = alpha * c1[v] + A[(size_t)r1 * HW + col];
        }
    }
}

// ---------------------------------------------------------------------------
extern "C" void kernel_launch(void* const* d_in, const int* in_sizes, int n_in,
                              void* d_out, int out_size, void* d_ws, size_t ws_size,
                              hipStream_t stream) {
    const float* x     = (const float*)d_in[0];
    const float* alpha = (const float*)d_in[1];
    float*       out   = (float*)d_out;

    float* partial = (float*)d_ws;                 // 128 * 1024 floats = 512 KB
    float* attn    = partial + 128 * TT * TT;      // 8 * 1024 floats   = 32 KB

    gram_kernel   <<<NB * 16, 256,  0, stream>>>(x, partial);
    softmax_kernel<<<NB,      1024, 0, stream>>>(partial, attn);
    apply_kernel  <<<NB * CC, 256,  0, stream>>>(x, attn, alpha, out);
}